// Vegas_69286412419533
// MI455X (gfx1250) — compile-verified
//
#include <hip/hip_runtime.h>
#include <stdint.h>

#define DIM    16
#define NINC   1000
#define NBINS  (DIM * NINC)        // 16000
#define NHIST  (2 * NBINS)         // 32000 floats = 128 KB
#define TPB    256
#define STAGEF (2 * TPB * DIM)     // double-buffer staging floats (32 KB)
#define LDSF   (NHIST + STAGEF)    // total dynamic LDS floats (160 KB)

// ---- CDNA5 async-to-LDS (device pass only; host pass sees 0) -------------
#if defined(__has_builtin)
#if __has_builtin(__builtin_amdgcn_global_load_async_to_lds_b128)
#define HAVE_ASYNC_LDS 1
#endif
#if __has_builtin(__builtin_amdgcn_s_wait_asynccnt)
#define HAVE_WAIT_ASYNC 1
#endif
#endif

#if defined(HAVE_ASYNC_LDS)
typedef int v4i __attribute__((vector_size(16)));
typedef __attribute__((address_space(1))) v4i* gv4i_p;
typedef __attribute__((address_space(3))) v4i* lv4i_p;

#if defined(HAVE_WAIT_ASYNC)
#define ASYNC_WAIT(n) __builtin_amdgcn_s_wait_asynccnt(n)
#else
#define ASYNC_WAIT(n) asm volatile("s_wait_asynccnt %0" ::"i"(n) : "memory")
#endif

// Stage one 64-byte u-row (4 x b128) from global into this thread's LDS slot.
__device__ __forceinline__ void async_copy_row(const float* gsrc, float* ldst) {
#pragma unroll
  for (int q = 0; q < 4; ++q)
    __builtin_amdgcn_global_load_async_to_lds_b128(
        (gv4i_p)(gsrc + 4 * q), (lv4i_p)(ldst + 4 * q), 0, 0);
}
#endif

// ---- Stage 0: seed output with the input sum_f / n_f base ----------------
__global__ void __launch_bounds__(TPB) vegas_init_out(const float* __restrict__ sumf_in,
                                                      const float* __restrict__ nf_in,
                                                      float* __restrict__ out) {
  int j = blockIdx.x * TPB + threadIdx.x;
  if (j < NBINS)      out[j] = sumf_in[j];
  else if (j < NHIST) out[j] = nf_in[j - NBINS];
}

// ---- Stage 1: LDS-privatized histogram, async double-buffered u stream ---
__global__ void __launch_bounds__(TPB) vegas_hist(const float* __restrict__ u,
                                                  const float* __restrict__ wgt,
                                                  const float* __restrict__ detj,
                                                  float* __restrict__ ws,
                                                  float* __restrict__ out,
                                                  long long batch, int nwg, int use_ws) {
  extern __shared__ float lds[];  // [NHIST hist][STAGEF staging] = 160 KB
  for (int i = threadIdx.x; i < NHIST; i += TPB) lds[i] = 0.0f;
  __syncthreads();

  float* lsum = lds;            // [DIM][NINC]
  float* lcnt = lds + NBINS;    // [DIM][NINC]
  const int tid = threadIdx.x;
  const long long stride = (long long)nwg * TPB;
  const long long base0  = (long long)blockIdx.x * TPB;

#if defined(HAVE_ASYNC_LDS)
  float* stage = lds + NHIST;   // 2 x [TPB][DIM]
  // Prologue: stage tile 0 (per-lane OOB guard).
  {
    long long b = base0 + tid;
    if (b < batch) async_copy_row(u + b * DIM, stage + tid * DIM);
  }
  int phase = 0;
  for (long long base = base0; base < batch; base += stride) {
    const long long b     = base + tid;
    const long long nbase = base + stride;
    // Issue next tile into the other buffer (per-lane guard for the tail).
    if (nbase < batch) {
      long long bn = nbase + tid;
      if (bn < batch)
        async_copy_row(u + bn * DIM, stage + (phase ^ 1) * (TPB * DIM) + tid * DIM);
    }
    // Overlap scalar stream loads with the async wait.
    float f = 0.0f;
    if (b < batch) { f = detj[b] * wgt[b]; f = fabsf(f * f); }
    // Block-uniform wait decision: next tile fully issued by every wave -> 4
    // older ops are exactly the current tile; otherwise drain completely.
    if (nbase + TPB <= batch) { ASYNC_WAIT(4); } else { ASYNC_WAIT(0); }

    if (b < batch) {
      const float4* row = (const float4*)(stage + phase * (TPB * DIM) + tid * DIM);
#pragma unroll
      for (int q = 0; q < 4; ++q) {
        float4 v = row[q];                       // ds_load_b128
        float xs[4] = {v.x, v.y, v.z, v.w};
#pragma unroll
        for (int c = 0; c < 4; ++c) {
          int d  = q * 4 + c;
          int iu = (int)(xs[c] * (float)NINC);   // trunc == floor for u >= 0
          iu = iu < 0 ? 0 : (iu > NINC - 1 ? NINC - 1 : iu);
          atomicAdd(&lsum[d * NINC + iu], f);    // ds_add_f32
          atomicAdd(&lcnt[d * NINC + iu], 1.0f); // ds_add_f32
        }
      }
    }
    phase ^= 1;
  }
#else
  // Fallback: direct coalesced global float4 loads.
  for (long long b = base0 + tid; b < batch; b += stride) {
    float f = detj[b] * wgt[b];
    f = fabsf(f * f);
    const float4* up = (const float4*)(u + b * DIM);
#pragma unroll
    for (int q = 0; q < 4; ++q) {
      float4 v = up[q];
      float xs[4] = {v.x, v.y, v.z, v.w};
#pragma unroll
      for (int c = 0; c < 4; ++c) {
        int d  = q * 4 + c;
        int iu = (int)(xs[c] * (float)NINC);
        iu = iu < 0 ? 0 : (iu > NINC - 1 ? NINC - 1 : iu);
        atomicAdd(&lsum[d * NINC + iu], f);
        atomicAdd(&lcnt[d * NINC + iu], 1.0f);
      }
    }
  }
#endif
  __syncthreads();

  if (use_ws) {
    // Private histogram -> per-WG row in scratch; reduced atomic-free in stage 2.
    float* dst = ws + (size_t)blockIdx.x * NHIST;
    for (int i = threadIdx.x; i < NHIST; i += TPB) dst[i] = lds[i];
  } else {
    for (int i = threadIdx.x; i < NHIST; i += TPB)
      unsafeAtomicAdd(&out[i], lds[i]);          // global_atomic_add_f32
  }
}

// ---- Stage 2: cross-workgroup reduction of scratch rows ------------------
__global__ void __launch_bounds__(TPB) vegas_reduce(const float* __restrict__ ws,
                                                    const float* __restrict__ sumf_in,
                                                    const float* __restrict__ nf_in,
                                                    float* __restrict__ out, int nwg) {
  int j = blockIdx.x * TPB + threadIdx.x;
  if (j >= NHIST) return;
  float acc = (j < NBINS) ? sumf_in[j] : nf_in[j - NBINS];
  const float* p = ws + j;
  for (int wg = 0; wg < nwg; ++wg) acc += p[(size_t)wg * NHIST];
  out[j] = acc;
}

extern "C" void kernel_launch(void* const* d_in, const int* in_sizes, int n_in,
                              void* d_out, int out_size, void* d_ws, size_t ws_size,
                              hipStream_t stream) {
  const float* u    = (const float*)d_in[0];
  const float* wgt  = (const float*)d_in[1];
  const float* detj = (const float*)d_in[2];
  const float* sumf = (const float*)d_in[3];
  const float* nf   = (const float*)d_in[4];
  float* out = (float*)d_out;

  const long long batch = (long long)in_sizes[1];  // weight element count

  // Deterministic config from ws_size: prefer the atomic-free two-stage path.
  const size_t per_wg = (size_t)NHIST * sizeof(float);  // 128 KB per WG row
  int nwg    = (int)(ws_size / per_wg);
  int use_ws = 1;
  if (nwg < 32) { use_ws = 0; nwg = 128; }   // tiny scratch: atomic flush, fewer WGs
  if (nwg > 512) nwg = 512;                  // cap stage-2 read volume (~64 MB)

  // Host pass can't see device __has_builtin: always size for hist + staging.
  const size_t shmem = (size_t)LDSF * sizeof(float);    // 160 KB dynamic LDS
  (void)hipFuncSetAttribute(reinterpret_cast<const void*>(vegas_hist),
                            hipFuncAttributeMaxDynamicSharedMemorySize, (int)shmem);

  vegas_init_out<<<(NHIST + TPB - 1) / TPB, TPB, 0, stream>>>(sumf, nf, out);
  vegas_hist<<<nwg, TPB, shmem, stream>>>(u, wgt, detj, (float*)d_ws, out,
                                          batch, nwg, use_ws);
  if (use_ws)
    vegas_reduce<<<(NHIST + TPB - 1) / TPB, TPB, 0, stream>>>((const float*)d_ws,
                                                              sumf, nf, out, nwg);
}